// MinkLoc3D_17703855194361
// MI455X (gfx1250) — compile-verified
//
#include <hip/hip_runtime.h>
#include <hip/hip_bf16.h>

// GeM pooling for MinkLoc3D:  clip(x,eps)^p -> segmented mean (B=16) -> ^(1/p) -> L2 norm.
// Memory-bound: 1.024 GB feature read @ 23.3 TB/s ~= 44 us floor.
// Segment-sum is done as onehot(batch)^T x x^p with V_WMMA_F32_16X16X4_F32:
// each wave keeps a 16(seg) x 16(col) f32 accumulator in 8 VGPRs across a 2048-row chunk.

typedef __attribute__((ext_vector_type(2))) float v2f;
typedef __attribute__((ext_vector_type(8))) float v8f;

#define GEM_C      256
#define GEM_B      16
#define CHUNK      2048
#define WG_THREADS 512   // 16 waves * 32 lanes -> 16 column tiles = 256 cols

// ---------------------------------------------------------------- zero scratch
__global__ __launch_bounds__(256)
void gem_zero(float* __restrict__ ws, int n) {
    int i = blockIdx.x * 256 + threadIdx.x;
    if (i < n) ws[i] = 0.0f;
}

// ------------------------------------------------- main WMMA segmented pow-sum
__global__ __launch_bounds__(WG_THREADS)
void gem_segsum_wmma(const float* __restrict__ feat,   // (N, 256) f32
                     const float* __restrict__ pptr,   // scalar p
                     const int*   __restrict__ bidx,   // (N,) sorted in [0,16)
                     float*       __restrict__ sums,   // ws: (16, 256) f32
                     int nrows)
{
    __shared__ int sid[CHUNK];

    const int tid    = threadIdx.x;
    const int chunk0 = blockIdx.x * CHUNK;

    // Stage this chunk's batch ids in LDS; out-of-range rows get id=-1 so their
    // one-hot row is all zero (keeps EXEC uniform for WMMA, no divergent loads).
    for (int i = tid; i < CHUNK; i += WG_THREADS) {
        int g  = chunk0 + i;
        sid[i] = (g < nrows) ? bidx[g] : -1;
    }
    __syncthreads();

    const int   lane = tid & 31;       // wave32
    const int   wave = tid >> 5;       // 0..15 -> which 16-column tile
    const int   half = lane >> 4;      // ISA 16x4 f32 A layout: lanes 0-15 K={0,1}
    const int   mn   = lane & 15;      //                        lanes16-31 K={2,3}
    const int   kb   = half * 2;       // K base held by this lane
    const int   col  = wave * 16 + mn; // feature column (B/D: N = lane&15)
    const float p    = pptr[0];

    v8f acc = {};                      // 16x16 f32 accumulator: [segment][col]

    for (int k = 0; k < CHUNK; k += 4) {
        // A (16x4 one-hot): A[m][kb+v] = (batch_id == m)
        int id0 = sid[k + kb + 0];
        int id1 = sid[k + kb + 1];
        v2f a;
        a.x = (id0 == mn) ? 1.0f : 0.0f;
        a.y = (id1 == mn) ? 1.0f : 0.0f;

        // B (4x16): x^p for two rows of this lane's column (clamped row for tail;
        // the zero one-hot kills any bogus contribution).
        int    r0  = chunk0 + k + kb;
        int    rc0 = min(r0,     nrows - 1);
        int    rc1 = min(r0 + 1, nrows - 1);
        float  x0  = feat[(size_t)rc0 * GEM_C + col];
        float  x1  = feat[(size_t)rc1 * GEM_C + col];
        v2f b;
        // x^p = exp2(p * log2(x));  v_log_f32/v_exp_f32 are base-2 on AMD.
        b.x = __builtin_amdgcn_exp2f(p * __builtin_amdgcn_logf(fmaxf(x0, 1e-6f)));
        b.y = __builtin_amdgcn_exp2f(p * __builtin_amdgcn_logf(fmaxf(x1, 1e-6f)));

        // D = A x B + C   (V_WMMA_F32_16X16X4_F32, full f32 precision)
        acc = __builtin_amdgcn_wmma_f32_16x16x4_f32(
                  false, a, false, b, (short)0, acc, false, false);

        // Stream-ahead prefetch into GL2 (gfx1250 global_prefetch_b8).
        int rp = min(r0 + 64, nrows - 1);
        __builtin_prefetch(&feat[(size_t)rp * GEM_C + col], 0, 1);
    }

    // Flush: C/D layout -> VGPR j holds segment M = j + half*8, column N = col.
    #pragma unroll
    for (int j = 0; j < 8; ++j) {
        int seg = j + half * 8;
        __hip_atomic_fetch_add(&sums[seg * GEM_C + col], acc[j],
                               __ATOMIC_RELAXED, __HIP_MEMORY_SCOPE_AGENT);
    }
}

// ------------------------------------------------------------------ counts
__global__ __launch_bounds__(256)
void gem_counts(const int* __restrict__ bidx, float* __restrict__ counts, int nrows)
{
    __shared__ float lc[GEM_B];
    int t = threadIdx.x;
    if (t < GEM_B) lc[t] = 0.0f;
    __syncthreads();

    int stride = gridDim.x * 256;
    for (int i = blockIdx.x * 256 + t; i < nrows; i += stride) {
        int b = bidx[i] & (GEM_B - 1);
        __hip_atomic_fetch_add(&lc[b], 1.0f,
                               __ATOMIC_RELAXED, __HIP_MEMORY_SCOPE_WORKGROUP);
    }
    __syncthreads();
    if (t < GEM_B)
        __hip_atomic_fetch_add(&counts[t], lc[t],
                               __ATOMIC_RELAXED, __HIP_MEMORY_SCOPE_AGENT);
}

// ---------------------------------------------------------------- finalize
__global__ __launch_bounds__(256)
void gem_finalize(const float* __restrict__ sums,    // (16,256)
                  const float* __restrict__ counts,  // (16,)
                  const float* __restrict__ pptr,
                  float*       __restrict__ out)     // (16,256)
{
    __shared__ float sdesc[GEM_B * GEM_C];
    __shared__ float red[GEM_C];
    const int   t    = threadIdx.x;                  // column 0..255
    const float invp = 1.0f / pptr[0];

    for (int r = 0; r < GEM_B; ++r) {
        float mean = sums[r * GEM_C + t] / counts[r];
        sdesc[r * GEM_C + t] =
            __builtin_amdgcn_exp2f(invp * __builtin_amdgcn_logf(mean));
    }
    __syncthreads();

    for (int r = 0; r < GEM_B; ++r) {
        float v = sdesc[r * GEM_C + t];
        red[t] = v * v;
        __syncthreads();
        for (int s = 128; s > 0; s >>= 1) {
            if (t < s) red[t] += red[t + s];
            __syncthreads();
        }
        float norm = sqrtf(red[0]);
        out[r * GEM_C + t] = v / fmaxf(norm, 1e-12f);
        __syncthreads();
    }
}

// ---------------------------------------------------------------- launcher
extern "C" void kernel_launch(void* const* d_in, const int* in_sizes, int n_in,
                              void* d_out, int out_size, void* d_ws, size_t ws_size,
                              hipStream_t stream) {
    const float* feat  = (const float*)d_in[0];   // (N, 256) f32
    const float* p     = (const float*)d_in[1];   // (1,) f32
    const int*   bidx  = (const int*)d_in[2];     // (N,) i32
    const int    nrows = in_sizes[2];

    float* ws     = (float*)d_ws;
    float* sums   = ws;                   // 16*256
    float* counts = ws + GEM_B * GEM_C;   // 16
    float* out    = (float*)d_out;

    const int wsElems = GEM_B * GEM_C + GEM_B;
    gem_zero<<<(wsElems + 255) / 256, 256, 0, stream>>>(ws, wsElems);

    const int nchunks = (nrows + CHUNK - 1) / CHUNK;
    gem_segsum_wmma<<<nchunks, WG_THREADS, 0, stream>>>(feat, p, bidx, sums, nrows);
    gem_counts<<<128, 256, 0, stream>>>(bidx, counts, nrows);
    gem_finalize<<<1, 256, 0, stream>>>(sums, counts, p, out);
}